// EGATlayer_87325275062875
// MI455X (gfx1250) — compile-verified
//
#include <hip/hip_runtime.h>
#include <hip/hip_bf16.h>

#define NNODE 2048
#define FIN   256
#define FOUT  256
#define NH    8

typedef __attribute__((ext_vector_type(2))) float v2f;
typedef __attribute__((ext_vector_type(8))) float v8f;

// ---------------------------------------------------------------------------
// qk[i,h] = (nf[i,:] . (Wq[:,h]+Wk[:,h])) * (1/16) / sqrt(8)
// ---------------------------------------------------------------------------
__global__ __launch_bounds__(256)
void qk_kernel(const float* __restrict__ nf, const float* __restrict__ Wq,
               const float* __restrict__ Wk, float* __restrict__ qk) {
    int t = blockIdx.x * blockDim.x + threadIdx.x;   // 0 .. NNODE*NH-1
    int i = t >> 3, h = t & 7;
    float acc = 0.f;
    for (int c = 0; c < FIN; ++c)
        acc += nf[i * FIN + c] * (Wq[c * NH + h] + Wk[c * NH + h]);
    qk[t] = acc * (0.0625f * 0.35355339059327373f);
}

// ---------------------------------------------------------------------------
// fp32 WMMA GEMM: one wave -> one 16x16 C tile. A is MxK row-major.
// TRANS_B=0: B is KxN row-major.  TRANS_B=1: B is NxK row-major (B^T GEMM).
// EPI: 0 = scale, 1 = elu, 2 = sigmoid(x)*(-dist)*deg
// ---------------------------------------------------------------------------
template <int TRANS_B, int EPI>
__global__ __launch_bounds__(32)
void gemm_wmma_f32(const float* __restrict__ A, const float* __restrict__ B,
                   float* __restrict__ C, int K, int lda, int ldb, int ldc,
                   float scale, const float* __restrict__ dist,
                   const float* __restrict__ deg) {
    const int lane = threadIdx.x & 31;
    const int half = lane >> 4;       // 0 for lanes 0-15, 1 for 16-31
    const int lm   = lane & 15;
    const int n0   = blockIdx.x * 16;
    const int m0   = blockIdx.y * 16;
    const int kA   = 2 * half;        // lane-half K offset within a 4-step

    const float* Arow = A + (size_t)(m0 + lm) * lda;
    const float* Brow = TRANS_B ? (B + (size_t)(n0 + lm) * ldb) : B;

    v8f acc = {};
    #pragma unroll 4
    for (int k = 0; k < K; k += 4) {
        v2f a, b;
        a.x = Arow[k + kA];
        a.y = Arow[k + kA + 1];
        if (TRANS_B) {
            b.x = Brow[k + kA];
            b.y = Brow[k + kA + 1];
        } else {
            b.x = B[(size_t)(k + kA) * ldb + n0 + lm];
            b.y = B[(size_t)(k + kA + 1) * ldb + n0 + lm];
        }
        acc = __builtin_amdgcn_wmma_f32_16x16x4_f32(
            false, a, false, b, (short)0, acc, false, false);
    }

    #pragma unroll
    for (int r = 0; r < 8; ++r) {
        int row = m0 + r + 8 * half;
        int col = n0 + lm;
        size_t idx = (size_t)row * ldc + col;
        float x = acc[r];
        if (EPI == 0) {
            C[idx] = x * scale;
        } else if (EPI == 1) {                       // elu
            C[idx] = x > 0.f ? x : (__expf(x) - 1.f);
        } else {                                     // conn epilogue
            float sig = 1.f / (1.f + __expf(-x));
            C[idx] = sig * (-dist[idx]) * deg[idx];
        }
    }
}

// ---------------------------------------------------------------------------
// Per-(column j, head h) online softmax stats over axis i.
// Block: 32 columns x 8 row-stripes. Coalesced single pass over edge arrays.
// ---------------------------------------------------------------------------
__global__ __launch_bounds__(256)
void colsoftmax_stats(const float* __restrict__ dist, const float* __restrict__ bond,
                      const float* __restrict__ deg, const float* __restrict__ qk,
                      const float* __restrict__ wb, const float* __restrict__ wbc,
                      const float* __restrict__ wg,
                      float* __restrict__ colmax, float* __restrict__ coldenom) {
    const float cb = wb[0] * wbc[0];
    const float cg = wg[0];
    const int tx = threadIdx.x;            // column within tile (0..31)
    const int ty = threadIdx.y;            // row stripe (0..7)
    const int j = blockIdx.x * 32 + tx;

    float m[NH], s[NH];
    #pragma unroll
    for (int h = 0; h < NH; ++h) { m[h] = -3.4e38f; s[h] = 0.f; }

    for (int i = ty; i < NNODE; i += 8) {
        size_t idx = (size_t)i * NNODE + j;
        float e = bond[idx] * cb + dist[idx] * cg;
        float mask = deg[idx] > 0.f ? 0.f : -1e9f;
        #pragma unroll
        for (int h = 0; h < NH; ++h) {
            float v = qk[i * NH + h] + e;
            float sc = (v > 0.f ? v : 0.2f * v) + mask;
            float mn = fmaxf(m[h], sc);
            s[h] = s[h] * __expf(m[h] - mn) + __expf(sc - mn);
            m[h] = mn;
        }
    }

    __shared__ float rm[NH][256], rs[NH][256];
    #pragma unroll
    for (int h = 0; h < NH; ++h) {
        rm[h][ty * 32 + tx] = m[h];
        rs[h][ty * 32 + tx] = s[h];
    }
    __syncthreads();

    // each thread merges the 8 stripe-partials for (j=tx tile col, h=ty)
    {
        int h = ty;
        float mm = rm[h][tx], ss = rs[h][tx];
        #pragma unroll
        for (int p = 1; p < 8; ++p) {
            float m2 = rm[h][p * 32 + tx], s2 = rs[h][p * 32 + tx];
            float mn = fmaxf(mm, m2);
            ss = ss * __expf(mm - mn) + s2 * __expf(m2 - mn);
            mm = mn;
        }
        int jj = blockIdx.x * 32 + tx;
        colmax[jj * NH + h]   = mm;
        coldenom[jj * NH + h] = ss;
    }
}

// ---------------------------------------------------------------------------
// attn_mean[i,j] = (1/8) * sum_h exp(s[i,j,h]-colmax[j,h]) / coldenom[j,h]
// ---------------------------------------------------------------------------
__global__ __launch_bounds__(256)
void attn_mean_kernel(const float* __restrict__ dist, const float* __restrict__ bond,
                      const float* __restrict__ deg, const float* __restrict__ qk,
                      const float* __restrict__ colmax, const float* __restrict__ coldenom,
                      const float* __restrict__ wb, const float* __restrict__ wbc,
                      const float* __restrict__ wg, float* __restrict__ attn) {
    const int j = blockIdx.x * 256 + threadIdx.x;
    const int i = blockIdx.y;
    const float cb = wb[0] * wbc[0];
    const float cg = wg[0];
    size_t idx = (size_t)i * NNODE + j;
    float e = bond[idx] * cb + dist[idx] * cg;
    float mask = deg[idx] > 0.f ? 0.f : -1e9f;
    float acc = 0.f;
    #pragma unroll
    for (int h = 0; h < NH; ++h) {
        float v = qk[i * NH + h] + e;
        float sc = (v > 0.f ? v : 0.2f * v) + mask;
        acc += __expf(sc - colmax[j * NH + h]) / coldenom[j * NH + h];
    }
    attn[idx] = acc * 0.125f;
}

// ---------------------------------------------------------------------------
// Row layernorm (mean/var over last axis, ddof=0), in place.
// ---------------------------------------------------------------------------
__global__ __launch_bounds__(256)
void rownorm_kernel(float* __restrict__ conn) {
    const int i = blockIdx.x;
    float sum = 0.f, sq = 0.f;
    for (int j = threadIdx.x; j < NNODE; j += 256) {
        float x = conn[(size_t)i * NNODE + j];
        sum += x; sq += x * x;
    }
    __shared__ float ssum[256], ssq[256];
    ssum[threadIdx.x] = sum; ssq[threadIdx.x] = sq;
    __syncthreads();
    for (int o = 128; o > 0; o >>= 1) {
        if (threadIdx.x < o) {
            ssum[threadIdx.x] += ssum[threadIdx.x + o];
            ssq[threadIdx.x]  += ssq[threadIdx.x + o];
        }
        __syncthreads();
    }
    float mean = ssum[0] * (1.f / NNODE);
    float var  = ssq[0] * (1.f / NNODE) - mean * mean;
    float inv  = rsqrtf(var + 1e-5f);
    for (int j = threadIdx.x; j < NNODE; j += 256) {
        size_t id = (size_t)i * NNODE + j;
        conn[id] = (conn[id] - mean) * inv;
    }
}

// ---------------------------------------------------------------------------
// out[i,j] = conn[i,j] + conn[j,i] via LDS 32x32 tiles (coalesced both ways)
// ---------------------------------------------------------------------------
__global__ __launch_bounds__(256)
void symadd_kernel(const float* __restrict__ conn, float* __restrict__ out) {
    __shared__ float tile[32][33];
    const int tx = threadIdx.x & 31;
    const int tg = threadIdx.x >> 5;      // 0..7
    const int j0 = blockIdx.x * 32, i0 = blockIdx.y * 32;
    for (int r = tg; r < 32; r += 8)
        tile[r][tx] = conn[(size_t)(j0 + r) * NNODE + i0 + tx];
    __syncthreads();
    for (int r = tg; r < 32; r += 8) {
        size_t id = (size_t)(i0 + r) * NNODE + j0 + tx;
        out[id] = conn[id] + tile[tx][r];
    }
}

// ---------------------------------------------------------------------------
extern "C" void kernel_launch(void* const* d_in, const int* in_sizes, int n_in,
                              void* d_out, int out_size, void* d_ws, size_t ws_size,
                              hipStream_t stream) {
    const float* nf   = (const float*)d_in[0];
    const float* dist = (const float*)d_in[1];
    const float* bond = (const float*)d_in[2];
    // d_in[3] = edge_direction: mathematically unused (only Y_0 couples scalars)
    const float* deg  = (const float*)d_in[4];
    const float* Wq   = (const float*)d_in[5];
    const float* Wk   = (const float*)d_in[6];
    const float* Wv   = (const float*)d_in[7];
    const float* wb   = (const float*)d_in[8];
    const float* wbc  = (const float*)d_in[9];
    const float* wg   = (const float*)d_in[10];

    float* out = (float*)d_out;
    float* un  = out;                               // [NNODE, FOUT]
    float* out_conn = out + (size_t)NNODE * FOUT;   // [NNODE, NNODE]

    // workspace layout (floats)
    float* ws = (float*)d_ws;
    float* qk       = ws;                      // NNODE*NH
    float* colmax   = qk + NNODE * NH;         // NNODE*NH
    float* coldenom = colmax + NNODE * NH;     // NNODE*NH
    float* v        = coldenom + NNODE * NH;   // NNODE*FOUT
    float* big      = v + (size_t)NNODE * FOUT; // NNODE*NNODE (attn, later conn)
    float* attn = big;
    float* conn = big;

    // 1) qk = nf @ (Wq+Wk) * (1/16)/sqrt(8)
    qk_kernel<<<(NNODE * NH) / 256, 256, 0, stream>>>(nf, Wq, Wk, qk);

    // 2) v = (nf @ Wv) * 1/16    (WMMA fp32, M=2048 N=256 K=256)
    {
        dim3 grid(FOUT / 16, NNODE / 16);
        gemm_wmma_f32<0, 0><<<grid, 32, 0, stream>>>(
            nf, Wv, v, FIN, FIN, FOUT, FOUT, 0.0625f, nullptr, nullptr);
    }

    // 3) per-(col j, head h) online softmax stats
    colsoftmax_stats<<<NNODE / 32, dim3(32, 8), 0, stream>>>(
        dist, bond, deg, qk, wb, wbc, wg, colmax, coldenom);

    // 4) attn_mean[i,j]
    attn_mean_kernel<<<dim3(NNODE / 256, NNODE), 256, 0, stream>>>(
        dist, bond, deg, qk, colmax, coldenom, wb, wbc, wg, attn);

    // 5) un = elu(attn_mean @ v)   (WMMA fp32, M=2048 N=256 K=2048)
    {
        dim3 grid(FOUT / 16, NNODE / 16);
        gemm_wmma_f32<0, 1><<<grid, 32, 0, stream>>>(
            attn, v, un, NNODE, NNODE, FOUT, FOUT, 1.f, nullptr, nullptr);
    }

    // 6) conn = sigmoid(un @ un^T) * (-dist) * deg   (WMMA fp32 NT, K=256)
    {
        dim3 grid(NNODE / 16, NNODE / 16);
        gemm_wmma_f32<1, 2><<<grid, 32, 0, stream>>>(
            un, un, conn, FOUT, FOUT, FOUT, NNODE, 1.f, dist, deg);
    }

    // 7) row layernorm in place
    rownorm_kernel<<<NNODE, 256, 0, stream>>>(conn);

    // 8) out_conn = conn + conn^T
    symadd_kernel<<<dim3(NNODE / 32, NNODE / 32), 256, 0, stream>>>(conn, out_conn);
}